// GNNCollaborativeFiltering_89481348645684
// MI455X (gfx1250) — compile-verified
//
#include <hip/hip_runtime.h>
#include <hip/hip_bf16.h>

typedef __attribute__((ext_vector_type(2))) float v2f;
typedef __attribute__((ext_vector_type(8))) float v8f;

#define N_USERS 250000
#define N_ITEMS 250000
#define N_NODES 500000
#define DIM 64

__device__ __forceinline__ void atomic_fadd(float* p, float v) {
    unsafeAtomicAdd(p, v);   // lowers to global_atomic_add_f32 on gfx1250
}

// ---------------- degree / normalization ----------------
__global__ void k_init_deg(float* deg, int n) {
    int i = blockIdx.x * blockDim.x + threadIdx.x;
    if (i < n) deg[i] = 1.0f;               // self-loop
}

__global__ void k_count_deg(const int* __restrict__ dst, float* deg, int E) {
    int e = blockIdx.x * blockDim.x + threadIdx.x;
    if (e < E) atomic_fadd(&deg[dst[e]], 1.0f);
}

__global__ void k_dinv(const float* __restrict__ deg, float* dinv, int n) {
    int i = blockIdx.x * blockDim.x + threadIdx.x;
    if (i < n) dinv[i] = rsqrtf(deg[i]);
}

// ---------------- embedding gather (x = concat of bag lookups) ----------------
__global__ void k_gather(const float* __restrict__ ut, const float* __restrict__ it,
                         const int* __restrict__ uid, const int* __restrict__ iid,
                         float* __restrict__ x) {
    int g = blockIdx.x * blockDim.x + threadIdx.x;   // over N_NODES*DIM
    if (g >= N_NODES * DIM) return;
    int n = g >> 6, k = g & 63;
    float v;
    if (n < N_USERS) v = ut[(long)uid[n] * DIM + k];
    else             v = it[(long)iid[n - N_USERS] * DIM + k];
    x[g] = v;
}

// ---------------- WMMA fp32 GEMM: H[M,64] = X[M,64] * W[64,64] ----------------
// 8 waves/block; wave w computes rows [row0, row0+16) across all 64 cols
// (4 accumulators of 16x16). K-loop: 16 steps of V_WMMA_F32_16X16X4_F32.
// W is staged in LDS pair-interleaved so each B fragment is one ds_load_b64:
//   lwp[(k>>1)*128 + col*2 + (k&1)] = W[k][col]
__global__ __launch_bounds__(256) void k_gemm64(const float* __restrict__ X,
                                                const float* __restrict__ Wg,
                                                float* __restrict__ H, int M) {
    __shared__ float lwp[64 * 64];
    for (int i = threadIdx.x; i < 64 * 64; i += 256) {
        int k = i >> 6, col = i & 63;
        lwp[((k >> 1) << 7) + (col << 1) + (k & 1)] = Wg[i];
    }
    __syncthreads();

    const int wave = threadIdx.x >> 5;
    const int lane = threadIdx.x & 31;
    const int mrow = lane & 15;
    const int khalf = lane >> 4;                 // 0: K={0,1}, 1: K={2,3} of each step
    const long row0 = ((long)blockIdx.x * 8 + wave) * 16;

    long r = row0 + mrow;
    if (r >= M) r = M - 1;                       // clamp reads; stores are guarded
    const float* xrow = X + r * DIM;

    v8f acc0 = {}, acc1 = {}, acc2 = {}, acc3 = {};
    for (int k0 = 0; k0 < 64; k0 += 4) {
        const int kb = k0 + 2 * khalf;           // this lane-half's K pair base
        // A fragment (16x4 f32): lane<16 -> K=k0,k0+1 ; lane>=16 -> K=k0+2,k0+3
        v2f a = *reinterpret_cast<const v2f*>(xrow + kb);
        // B fragments (4x16 f32): one aligned 8B LDS load per 16-col block
        const float* bp = lwp + ((kb >> 1) << 7) + (mrow << 1);
        v2f b0 = *reinterpret_cast<const v2f*>(bp +  0);
        v2f b1 = *reinterpret_cast<const v2f*>(bp + 32);
        v2f b2 = *reinterpret_cast<const v2f*>(bp + 64);
        v2f b3 = *reinterpret_cast<const v2f*>(bp + 96);
        acc0 = __builtin_amdgcn_wmma_f32_16x16x4_f32(false, a, false, b0, (short)0, acc0, false, false);
        acc1 = __builtin_amdgcn_wmma_f32_16x16x4_f32(false, a, false, b1, (short)0, acc1, false, false);
        acc2 = __builtin_amdgcn_wmma_f32_16x16x4_f32(false, a, false, b2, (short)0, acc2, false, false);
        acc3 = __builtin_amdgcn_wmma_f32_16x16x4_f32(false, a, false, b3, (short)0, acc3, false, false);
    }
    // C/D layout: VGPR j -> M = j + 8*khalf, N = mrow (per 16-col block)
    #pragma unroll
    for (int j = 0; j < 8; ++j) {
        long rr = row0 + j + 8 * khalf;
        if (rr < M) {
            float* hp = H + rr * DIM + mrow;
            hp[0]  = acc0[j];
            hp[16] = acc1[j];
            hp[32] = acc2[j];
            hp[48] = acc3[j];
        }
    }
}

// ---------------- self-loop message (also initializes the accumulator) ----
__global__ void k_selfloop(const float* __restrict__ h, const float* __restrict__ dinv,
                           float* __restrict__ out) {
    int g = blockIdx.x * blockDim.x + threadIdx.x;   // over N_NODES*DIM
    if (g >= N_NODES * DIM) return;
    int n = g >> 6;
    float di = dinv[n];
    out[g] = h[g] * di * di;
}

// ---------------- edge scatter: out[d] += h[s] * dinv[s]*dinv[d] ---------------
__global__ void k_scatter(const int* __restrict__ src, const int* __restrict__ dst,
                          const float* __restrict__ h, const float* __restrict__ dinv,
                          float* __restrict__ out, int E) {
    long g = (long)blockIdx.x * blockDim.x + threadIdx.x;  // 64 threads per edge
    if (g >= (long)E * DIM) return;
    int e = (int)(g >> 6), k = (int)(g & 63);
    int s = src[e], d = dst[e];
    float norm = dinv[s] * dinv[d];
    atomic_fadd(&out[(long)d * DIM + k], h[(long)s * DIM + k] * norm);
}

// ---------------- bias + ReLU (in place) ----------------
__global__ void k_bias_relu(float* __restrict__ x, const float* __restrict__ bias) {
    int g = blockIdx.x * blockDim.x + threadIdx.x;   // over N_NODES*DIM
    if (g >= N_NODES * DIM) return;
    float v = x[g] + bias[g & 63];
    x[g] = v > 0.0f ? v : 0.0f;
}

// ---------------- final prediction: one wave32 per row ----------------
__global__ __launch_bounds__(256) void k_predict(const float* __restrict__ X,
                                                 const int* __restrict__ uid,
                                                 const int* __restrict__ iid,
                                                 const float* __restrict__ Wp,
                                                 const float* __restrict__ bp,
                                                 float* __restrict__ out, int B) {
    int wave = threadIdx.x >> 5, lane = threadIdx.x & 31;
    int row = blockIdx.x * 8 + wave;
    if (row >= B) return;                        // wave-uniform exit
    const float* xu = X + (long)uid[row] * DIM;
    const float* xi = X + (long)(N_USERS + iid[row]) * DIM;
    int c = 2 * lane;
    float sum = xu[c] * Wp[c] + xu[c + 1] * Wp[c + 1]
              + xi[c] * Wp[DIM + c] + xi[c + 1] * Wp[DIM + c + 1];
    #pragma unroll
    for (int off = 16; off > 0; off >>= 1) sum += __shfl_down(sum, off, 32);
    if (lane == 0) out[row] = sum + bp[0];
}

extern "C" void kernel_launch(void* const* d_in, const int* in_sizes, int n_in,
                              void* d_out, int out_size, void* d_ws, size_t ws_size,
                              hipStream_t stream) {
    const float* user_table = (const float*)d_in[0];
    const float* item_table = (const float*)d_in[1];
    const float* W1 = (const float*)d_in[2];
    const float* b1 = (const float*)d_in[3];
    const float* W2 = (const float*)d_in[4];
    const float* b2 = (const float*)d_in[5];
    const float* Wp = (const float*)d_in[6];
    const float* bp = (const float*)d_in[7];
    const int*   edges = (const int*)d_in[8];
    const int*   uid = (const int*)d_in[9];
    const int*   iid = (const int*)d_in[10];

    const int E = in_sizes[8] / 2;
    const int B = in_sizes[9];
    const int* src = edges;
    const int* dst = edges + E;

    const size_t XB = (size_t)N_NODES * DIM * sizeof(float);   // 128 MB
    char* ws = (char*)d_ws;
    float* xa   = (float*)(ws);
    float* h    = (float*)(ws + XB);
    float* xb   = (float*)(ws + 2 * XB);
    float* deg  = (float*)(ws + 3 * XB);
    float* dinv = (float*)(ws + 3 * XB + (size_t)N_NODES * sizeof(float));

    const int T = 256;
    const int gNodes = (N_NODES + T - 1) / T;
    const int gElems = (N_NODES * DIM + T - 1) / T;
    const int gEdges = (E + T - 1) / T;
    const int gScat  = (int)(((long)E * DIM + T - 1) / T);
    const int gGemm  = (N_NODES + 127) / 128;
    const int gPred  = (B + 7) / 8;

    // degrees + symmetric normalization
    k_init_deg<<<gNodes, T, 0, stream>>>(deg, N_NODES);
    k_count_deg<<<gEdges, T, 0, stream>>>(dst, deg, E);
    k_dinv<<<gNodes, T, 0, stream>>>(deg, dinv, N_NODES);

    // x = concat(user_table[user_ids], item_table[item_ids])
    k_gather<<<gElems, T, 0, stream>>>(user_table, item_table, uid, iid, xa);

    // layer 1: xa -> h -> xb
    k_gemm64<<<gGemm, T, 0, stream>>>(xa, W1, h, N_NODES);
    k_selfloop<<<gElems, T, 0, stream>>>(h, dinv, xb);
    k_scatter<<<gScat, T, 0, stream>>>(src, dst, h, dinv, xb, E);
    k_bias_relu<<<gElems, T, 0, stream>>>(xb, b1);

    // layer 2: xb -> h -> xa
    k_gemm64<<<gGemm, T, 0, stream>>>(xb, W2, h, N_NODES);
    k_selfloop<<<gElems, T, 0, stream>>>(h, dinv, xa);
    k_scatter<<<gScat, T, 0, stream>>>(src, dst, h, dinv, xa, E);
    k_bias_relu<<<gElems, T, 0, stream>>>(xa, b2);

    // prediction head
    k_predict<<<gPred, T, 0, stream>>>(xa, uid, iid, Wp, bp, (float*)d_out, B);
}